// GraphAttentionLayer_8203387535412
// MI455X (gfx1250) — compile-verified
//
#include <hip/hip_runtime.h>

#define BATCH 8
#define NNODE 4096
#define NEDGE 32768
#define DIM   256
#define EDIM  64
#define NHEAD 8
#define HDIM  32

typedef __attribute__((ext_vector_type(2))) float v2f;
typedef __attribute__((ext_vector_type(8))) float v8f;
typedef __attribute__((ext_vector_type(4))) int   v4i;
typedef __attribute__((address_space(1))) v4i gv4i;   // global int4
typedef __attribute__((address_space(3))) v4i lv4i;   // LDS int4

__device__ __forceinline__ v8f wmma4(v2f a, v2f b, v8f c) {
  // D = A(16x4,f32) * B(4x16,f32) + C(16x16,f32) on the CDNA5 matrix pipe
  return __builtin_amdgcn_wmma_f32_16x16x4_f32(false, a, false, b, (short)0, c,
                                               false, false);
}

// ---- async global->LDS staging (GLOBAL_LOAD_ASYNC_TO_LDS_B128, ASYNCcnt) ----
#if defined(__gfx1250__) &&                                        \
    __has_builtin(__builtin_amdgcn_global_load_async_to_lds_b128) && \
    __has_builtin(__builtin_amdgcn_s_wait_asynccnt)
#define HAVE_ASYNC 1
#else
#define HAVE_ASYNC 0
#endif

__device__ __forceinline__ void async_cp16(const float* gsrc, float* ldst) {
#if HAVE_ASYNC
  __builtin_amdgcn_global_load_async_to_lds_b128(
      (gv4i*)(unsigned long long)gsrc, (lv4i*)ldst, 0, 0);
#else
  *(float4*)ldst = *(const float4*)gsrc;
#endif
}

__device__ __forceinline__ void async_wait0() {
#if HAVE_ASYNC
  __builtin_amdgcn_s_wait_asynccnt(0);
#endif
}

// ---------------------------------------------------------------- init ws
__global__ __launch_bounds__(256) void k_init(float* __restrict__ MX,
                                              float* __restrict__ DEN,
                                              float* __restrict__ AGG) {
  int i = blockIdx.x * 256 + threadIdx.x;            // covers B*N*D
  AGG[i] = 0.f;
  if (i < BATCH * NNODE * NHEAD) { MX[i] = -INFINITY; DEN[i] = 0.f; }
}

// --------------------- Q = X*Wq, K = X*Wk (WMMA + double-buffered async LDS)
__global__ __launch_bounds__(256) void k_qk(const float* __restrict__ nf,
                                            const float* __restrict__ Wq,
                                            const float* __restrict__ Wk,
                                            float* __restrict__ Q,
                                            float* __restrict__ K) {
  __shared__ float Als[2][128 * 36];   // 128 rows x 32 k, padded stride 36
  __shared__ float Bqs[2][32 * 64];
  __shared__ float Bks[2][32 * 64];
  const int tid  = threadIdx.x;
  const int wave = tid >> 5, lane = tid & 31;
  const int ln   = lane & 15, koff = (lane >> 4) * 2;
  const int mblk   = blockIdx.x >> 2;        // 256 row-blocks of 128
  const int nchunk = (blockIdx.x & 3) * 64;  // 4 column chunks of 64
  const int mrow0  = mblk * 128;
  const int mbase  = wave * 16;

  auto load_panel = [&](int kp, int buf) {
    for (int it = 0; it < 4; ++it) {          // A panel: 1024 x b128
      int lin = it * 256 + tid;
      int row = lin >> 3, c4 = lin & 7;
      async_cp16(nf + (mrow0 + row) * DIM + kp + c4 * 4,
                 &Als[buf][row * 36 + c4 * 4]);
    }
    for (int it = 0; it < 2; ++it) {          // 2 weight panels: 512 x b128
      int lin = it * 256 + tid;
      int row = lin >> 4, c4 = lin & 15;
      async_cp16(Wq + (kp + row) * DIM + nchunk + c4 * 4,
                 &Bqs[buf][row * 64 + c4 * 4]);
      async_cp16(Wk + (kp + row) * DIM + nchunk + c4 * 4,
                 &Bks[buf][row * 64 + c4 * 4]);
    }
  };

  v8f aq[4], ak[4];
  for (int t = 0; t < 4; ++t)
    for (int r = 0; r < 8; ++r) { aq[t][r] = 0.f; ak[t][r] = 0.f; }

  load_panel(0, 0);
  async_wait0();
  __syncthreads();

  for (int kp = 0; kp < DIM; kp += 32) {
    int buf = (kp >> 5) & 1;
    if (kp + 32 < DIM) load_panel(kp + 32, buf ^ 1);  // prefetch next panel
    for (int kk = 0; kk < 32; kk += 4) {
      v2f a = *(const v2f*)(&Als[buf][(mbase + ln) * 36 + kk + koff]);
      int br0 = (kk + koff) * 64;
      for (int t = 0; t < 4; ++t) {
        int c = t * 16 + ln;
        v2f bq; bq[0] = Bqs[buf][br0 + c]; bq[1] = Bqs[buf][br0 + 64 + c];
        v2f bk; bk[0] = Bks[buf][br0 + c]; bk[1] = Bks[buf][br0 + 64 + c];
        aq[t] = wmma4(a, bq, aq[t]);
        ak[t] = wmma4(a, bk, ak[t]);
      }
    }
    async_wait0();
    __syncthreads();
  }
  for (int t = 0; t < 4; ++t) {
    int col = nchunk + t * 16 + ln;
    for (int r = 0; r < 8; ++r) {
      int grow = mrow0 + mbase + r + ((lane >> 4) << 3);
      Q[grow * DIM + col] = aq[t][r];
      K[grow * DIM + col] = ak[t][r];
    }
  }
}

// ------------------------------- edge attention logits + scatter-max (fp32)
__global__ __launch_bounds__(256) void k_logits(
    const float* __restrict__ Q, const float* __restrict__ Kt,
    const float* __restrict__ ef, const int* __restrict__ eidx,
    const unsigned char* __restrict__ lmask, const float* __restrict__ Wedge,
    float* __restrict__ EX, float* __restrict__ MX) {
  __shared__ float We[EDIM * NHEAD];
  int tid = threadIdx.x;
  for (int i = tid; i < EDIM * NHEAD; i += 256) We[i] = Wedge[i];
  __syncthreads();
  int g = blockIdx.x * 256 + tid;           // (b, e, h) flat
  int b = g >> 18;                          // NEDGE*NHEAD = 2^18
  int eg = g >> 3;                          // b*NEDGE + e
  int h = g & 7;
  int s = eidx[eg * 2 + 0];
  int d = eidx[eg * 2 + 1];
  const float4* q4 = (const float4*)(Q  + (b * NNODE + d) * DIM + h * HDIM);
  const float4* k4 = (const float4*)(Kt + (b * NNODE + s) * DIM + h * HDIM);
  float qk = 0.f;
  for (int i = 0; i < 8; ++i) {
    float4 a = q4[i], c = k4[i];
    qk += a.x * c.x + a.y * c.y + a.z * c.z + a.w * c.w;
  }
  const float4* e4 = (const float4*)(ef + eg * EDIM);
  float ed = 0.f;
  for (int i = 0; i < 16; ++i) {
    float4 v = e4[i];
    ed += v.x * We[(i * 4 + 0) * NHEAD + h] + v.y * We[(i * 4 + 1) * NHEAD + h]
        + v.z * We[(i * 4 + 2) * NHEAD + h] + v.w * We[(i * 4 + 3) * NHEAD + h];
  }
  float attn = qk * 0.17677669529663687f + ed;  // 1/sqrt(32)
  if (!lmask[eg]) attn = -INFINITY;
  EX[g] = attn;
  __hip_atomic_fetch_max(&MX[(b * NNODE + d) * NHEAD + h], attn,
                         __ATOMIC_RELAXED, __HIP_MEMORY_SCOPE_AGENT);
}

// ------------------------------------------- exp(attn - max) + denom scatter
__global__ __launch_bounds__(256) void k_expden(const int* __restrict__ eidx,
                                                const float* __restrict__ MX,
                                                float* __restrict__ EX,
                                                float* __restrict__ DEN) {
  int g = blockIdx.x * 256 + threadIdx.x;
  int b = g >> 18, eg = g >> 3, h = g & 7;
  int d = eidx[eg * 2 + 1];
  int mi = (b * NNODE + d) * NHEAD + h;
  float mx = fmaxf(MX[mi], -1e6f);
  float ex = __expf(EX[g] - mx);
  EX[g] = ex;
  atomicAdd(&DEN[mi], ex);
}

// -- V = [nf[src]||ef] * Wv (WMMA, double-buffered async Wv panels),
//    apply softmax weights, dual scatter-add to AGG[dst] and AGG[src]
__global__ __launch_bounds__(256) void k_vmsg(
    const float* __restrict__ nf, const float* __restrict__ ef,
    const int* __restrict__ eidx, const float* __restrict__ Wv,
    const float* __restrict__ EX, const float* __restrict__ DEN,
    float* __restrict__ AGG) {
  __shared__ float Bls[2][32 * 256];         // double-buffered Wv k-panels
  const int tid = threadIdx.x;
  const int wave = tid >> 5, lane = tid & 31;
  const int ln = lane & 15, koff = (lane >> 4) * 2;
  const int b = blockIdx.x >> 8;             // 256 blocks per batch
  const int eg0 = blockIdx.x * 128 + wave * 16;
  const int egm = eg0 + ln;
  const int sm = eidx[egm * 2];
  const float* arow = nf + (b * NNODE + sm) * DIM;
  const float* erow = ef + egm * EDIM;

  auto load_panel = [&](int kp, int buf) {
    for (int it = 0; it < 8; ++it) {         // 32x256 panel = 2048 x b128
      int lin = it * 256 + tid;
      int row = lin >> 6, c4 = lin & 63;
      async_cp16(Wv + (kp + row) * DIM + c4 * 4, &Bls[buf][row * 256 + c4 * 4]);
    }
  };

  v8f acc[16];
  for (int t = 0; t < 16; ++t)
    for (int r = 0; r < 8; ++r) acc[t][r] = 0.f;

  load_panel(0, 0);
  async_wait0();
  __syncthreads();

  for (int kp = 0; kp < DIM + EDIM; kp += 32) {
    int buf = (kp >> 5) & 1;
    if (kp + 32 < DIM + EDIM) load_panel(kp + 32, buf ^ 1);
    for (int kk = 0; kk < 32; kk += 4) {
      int kg = kp + kk + koff;
      const float* ap = (kg < DIM) ? (arow + kg) : (erow + (kg - DIM));
      v2f a = *(const v2f*)ap;
      int br0 = (kk + koff) * 256;
      for (int t = 0; t < 16; ++t) {
        int c = t * 16 + ln;
        v2f bb; bb[0] = Bls[buf][br0 + c]; bb[1] = Bls[buf][br0 + 256 + c];
        acc[t] = wmma4(a, bb, acc[t]);
      }
    }
    async_wait0();
    __syncthreads();
  }
  for (int r = 0; r < 8; ++r) {
    int m = r + ((lane >> 4) << 3);
    int eg = eg0 + m;
    int s = eidx[eg * 2], d = eidx[eg * 2 + 1];
    int nb = b * NNODE;
    float w[8];
    for (int h = 0; h < 8; ++h) {
      float den = DEN[(nb + d) * NHEAD + h];
      w[h] = EX[eg * NHEAD + h] / fmaxf(den, 1e-8f);
    }
    float* dstp = AGG + (nb + d) * DIM;
    float* srcp = AGG + (nb + s) * DIM;
    for (int t = 0; t < 16; ++t) {
      int c = t * 16 + ln;
      float v = acc[t][r] * w[t >> 1];       // head = n-tile/2 (16 cols/tile)
      atomicAdd(dstp + c, v);
      atomicAdd(srcp + c, v);
    }
  }
}

// ----------------- out = AGG*Wout + b; residual; LayerNorm; node mask (WMMA)
__global__ __launch_bounds__(128) void k_outln(
    const float* __restrict__ AGG, const float* __restrict__ nf,
    const float* __restrict__ Wout, const float* __restrict__ bout,
    const float* __restrict__ gamma, const float* __restrict__ beta,
    const unsigned char* __restrict__ nmask, float* __restrict__ out) {
  __shared__ float xt[16 * 256];
  const int tid = threadIdx.x;
  const int wave = tid >> 5, lane = tid & 31;
  const int ln = lane & 15, koff = (lane >> 4) * 2;
  const int row0 = blockIdx.x * 16;
  const float* arow = AGG + (row0 + ln) * DIM;

  v8f acc[4];
  for (int j = 0; j < 4; ++j)
    for (int r = 0; r < 8; ++r) acc[j][r] = 0.f;

  for (int k = 0; k < DIM; k += 4) {
    v2f a = *(const v2f*)(arow + k + koff);
    int br0 = (k + koff) * DIM;
    for (int j = 0; j < 4; ++j) {
      int c = (wave * 4 + j) * 16 + ln;
      v2f bb; bb[0] = Wout[br0 + c]; bb[1] = Wout[br0 + DIM + c];
      acc[j] = wmma4(a, bb, acc[j]);
    }
  }
  for (int j = 0; j < 4; ++j) {
    int c = (wave * 4 + j) * 16 + ln;
    for (int r = 0; r < 8; ++r) {
      int m = r + ((lane >> 4) << 3);
      xt[m * 256 + c] = acc[j][r] + bout[c] + nf[(row0 + m) * DIM + c];
    }
  }
  __syncthreads();
  for (int i = 0; i < 4; ++i) {
    int m = wave * 4 + i;
    float s = 0.f, s2 = 0.f;
    for (int j = 0; j < 8; ++j) {
      float x = xt[m * 256 + lane * 8 + j];
      s += x; s2 += x * x;
    }
    for (int off = 16; off > 0; off >>= 1) {   // wave32 reduction
      s  += __shfl_xor(s,  off, 32);
      s2 += __shfl_xor(s2, off, 32);
    }
    float mean = s * (1.f / 256.f);
    float var  = fmaxf(s2 * (1.f / 256.f) - mean * mean, 0.f);
    float rstd = rsqrtf(var + 1e-5f);
    int grow = row0 + m;
    float msk = nmask[grow] ? 1.f : 0.f;
    for (int j = 0; j < 8; ++j) {
      int c = lane * 8 + j;
      float x = xt[m * 256 + c];
      out[grow * DIM + c] = ((x - mean) * rstd * gamma[c] + beta[c]) * msk;
    }
  }
}

extern "C" void kernel_launch(void* const* d_in, const int* in_sizes, int n_in,
                              void* d_out, int out_size, void* d_ws,
                              size_t ws_size, hipStream_t stream) {
  const float* nf    = (const float*)d_in[0];
  const float* ef    = (const float*)d_in[1];
  const int*   eidx  = (const int*)d_in[2];
  const unsigned char* nmask = (const unsigned char*)d_in[3];
  const unsigned char* lmask = (const unsigned char*)d_in[4];
  const float* Wq    = (const float*)d_in[5];
  const float* Wk    = (const float*)d_in[6];
  const float* Wv    = (const float*)d_in[7];
  const float* Wedge = (const float*)d_in[8];
  const float* Wout  = (const float*)d_in[9];
  const float* bout  = (const float*)d_in[10];
  const float* gamma = (const float*)d_in[11];
  const float* beta  = (const float*)d_in[12];
  float* out = (float*)d_out;

  float* Q   = (float*)d_ws;                       // [B*N*D]
  float* K   = Q   + BATCH * NNODE * DIM;          // [B*N*D]
  float* EX  = K   + BATCH * NNODE * DIM;          // [B*E*H] logits->exp
  float* MX  = EX  + BATCH * NEDGE * NHEAD;        // [B*N*H]
  float* DEN = MX  + BATCH * NNODE * NHEAD;        // [B*N*H]
  float* AGG = DEN + BATCH * NNODE * NHEAD;        // [B*N*D]

  k_init  <<<BATCH * NNODE * DIM / 256, 256, 0, stream>>>(MX, DEN, AGG);
  k_qk    <<<(BATCH * NNODE / 128) * (DIM / 64), 256, 0, stream>>>(nf, Wq, Wk, Q, K);
  k_logits<<<BATCH * NEDGE * NHEAD / 256, 256, 0, stream>>>(Q, K, ef, eidx, lmask,
                                                            Wedge, EX, MX);
  k_expden<<<BATCH * NEDGE * NHEAD / 256, 256, 0, stream>>>(eidx, MX, EX, DEN);
  k_vmsg  <<<BATCH * NEDGE / 128, 256, 0, stream>>>(nf, ef, eidx, Wv, EX, DEN, AGG);
  k_outln <<<BATCH * NNODE / 16, 128, 0, stream>>>(AGG, nf, Wout, bout, gamma,
                                                   beta, nmask, out);
}